// PolynomialLearningRulePair_69269232550109
// MI455X (gfx1250) — compile-verified
//
#include <hip/hip_runtime.h>
#include <hip/hip_bf16.h>

// ---------------------------------------------------------------------------
// Types for WMMA f32 16x16x4 (wave32): A,B = 2 floats/lane; C,D = 8 floats/lane
// ---------------------------------------------------------------------------
typedef float v2f __attribute__((ext_vector_type(2)));
typedef float v8f __attribute__((ext_vector_type(8)));

// ---------------------------------------------------------------------------
// Compile-time exponent tables (mirrors itertools.product order, sum<=3)
//   W blocks : 84 distinct (e0..e5) tuples ; terms: 120 (block, e6) pairs
//   M blocks : 35 distinct (e0..e3) tuples ; terms:  56 (block, e4) pairs
//   B combos : 10 distinct (e0,e1)         ; terms:  20 (comb, e2) pairs
// ---------------------------------------------------------------------------
struct Tables {
  signed char wblkA[84][4];  // exponents for (s1, y1, s2, y2) = (e1,e2,e4,e5)
  signed char wblkB[84][2];  // exponents for (x1, x2)         = (e0,e3)
  short       wterm_blk[120];
  signed char wterm_e[120];  // e6 (w power)

  signed char mblkA[35][2];  // left  = y1^e1 * y2^e3
  signed char mblkB[35][2];  // right = y1^e0 * y2^e2
  short       mterm_blk[56];
  signed char mterm_e[56];   // e4 (m power)

  signed char bcomb[10][2];  // (e0,e1) for y1^e0*y2^e1 column sums
  short       bterm_c[20];
  signed char bterm_e[20];   // e2 (b power)
};

constexpr Tables makeTables() {
  Tables t{};
  // ---- W blocks: enumerate (e0..e5), rightmost fastest (product order) ----
  int nwb = 0;
  for (int e0 = 0; e0 < 4; ++e0)
    for (int e1 = 0; e1 < 4; ++e1)
      for (int e2 = 0; e2 < 4; ++e2)
        for (int e3 = 0; e3 < 4; ++e3)
          for (int e4 = 0; e4 < 4; ++e4)
            for (int e5 = 0; e5 < 4; ++e5)
              if (e0 + e1 + e2 + e3 + e4 + e5 <= 3) {
                t.wblkA[nwb][0] = (signed char)e1;
                t.wblkA[nwb][1] = (signed char)e2;
                t.wblkA[nwb][2] = (signed char)e4;
                t.wblkA[nwb][3] = (signed char)e5;
                t.wblkB[nwb][0] = (signed char)e0;
                t.wblkB[nwb][1] = (signed char)e3;
                ++nwb;
              }
  // ---- W terms: enumerate (e0..e6), rightmost fastest ----
  int nwt = 0;
  for (int e0 = 0; e0 < 4; ++e0)
    for (int e1 = 0; e1 < 4; ++e1)
      for (int e2 = 0; e2 < 4; ++e2)
        for (int e3 = 0; e3 < 4; ++e3)
          for (int e4 = 0; e4 < 4; ++e4)
            for (int e5 = 0; e5 < 4; ++e5)
              for (int e6 = 0; e6 < 4; ++e6)
                if (e0 + e1 + e2 + e3 + e4 + e5 + e6 <= 3) {
                  int bi = -1;
                  for (int j = 0; j < nwb; ++j)
                    if (t.wblkB[j][0] == e0 && t.wblkA[j][0] == e1 &&
                        t.wblkA[j][1] == e2 && t.wblkB[j][1] == e3 &&
                        t.wblkA[j][2] == e4 && t.wblkA[j][3] == e5) { bi = j; break; }
                  t.wterm_blk[nwt] = (short)bi;
                  t.wterm_e[nwt]   = (signed char)e6;
                  ++nwt;
                }
  // ---- M blocks: enumerate (e0..e3) ----
  int nmb = 0;
  for (int e0 = 0; e0 < 4; ++e0)
    for (int e1 = 0; e1 < 4; ++e1)
      for (int e2 = 0; e2 < 4; ++e2)
        for (int e3 = 0; e3 < 4; ++e3)
          if (e0 + e1 + e2 + e3 <= 3) {
            t.mblkA[nmb][0] = (signed char)e1;  // left:  y1^e1
            t.mblkA[nmb][1] = (signed char)e3;  //        y2^e3
            t.mblkB[nmb][0] = (signed char)e0;  // right: y1^e0
            t.mblkB[nmb][1] = (signed char)e2;  //        y2^e2
            ++nmb;
          }
  // ---- M terms: enumerate (e0..e4) ----
  int nmt = 0;
  for (int e0 = 0; e0 < 4; ++e0)
    for (int e1 = 0; e1 < 4; ++e1)
      for (int e2 = 0; e2 < 4; ++e2)
        for (int e3 = 0; e3 < 4; ++e3)
          for (int e4 = 0; e4 < 4; ++e4)
            if (e0 + e1 + e2 + e3 + e4 <= 3) {
              int bi = -1;
              for (int j = 0; j < nmb; ++j)
                if (t.mblkB[j][0] == e0 && t.mblkA[j][0] == e1 &&
                    t.mblkB[j][1] == e2 && t.mblkA[j][1] == e3) { bi = j; break; }
              t.mterm_blk[nmt] = (short)bi;
              t.mterm_e[nmt]   = (signed char)e4;
              ++nmt;
            }
  // ---- B combos + terms ----
  int nbc = 0;
  for (int e0 = 0; e0 < 4; ++e0)
    for (int e1 = 0; e1 < 4; ++e1)
      if (e0 + e1 <= 3) {
        t.bcomb[nbc][0] = (signed char)e0;
        t.bcomb[nbc][1] = (signed char)e1;
        ++nbc;
      }
  int nbt = 0;
  for (int e0 = 0; e0 < 4; ++e0)
    for (int e1 = 0; e1 < 4; ++e1)
      for (int e2 = 0; e2 < 4; ++e2)
        if (e0 + e1 + e2 <= 3) {
          int ci = -1;
          for (int j = 0; j < nbc; ++j)
            if (t.bcomb[j][0] == e0 && t.bcomb[j][1] == e1) { ci = j; break; }
          t.bterm_c[nbt] = (short)ci;
          t.bterm_e[nbt] = (signed char)e2;
          ++nbt;
        }
  return t;
}

__constant__ Tables c_tab = makeTables();

#define NW_BLOCKS 84
#define NM_BLOCKS 35
#define N_BLOCKS  119
#define BTOT      65536
#define KCHUNKS   16
#define KCHUNK    (BTOT / KCHUNKS)   // 4096 rows per workgroup
#define KSUB      64                 // LDS-staged sub-chunk
#define LDS_STRIDE 68                // 64 + 4 pad: conflict-free b64 fragment loads
#define THREADS   128                // 4 wave32 = one 32x32 super-tile each

// x^e for e in 0..3 ; skips the load entirely when e==0 (uniform branch)
__device__ __forceinline__ float powld(const float* __restrict__ p, int idx, int e) {
  if (e == 0) return 1.0f;
  float v = p[idx];
  float r = v;
  if (e > 1) r *= v;
  if (e > 2) r *= v;
  return r;
}

__device__ __forceinline__ v8f wmma_f32(v2f a, v2f b, v8f c) {
  return __builtin_amdgcn_wmma_f32_16x16x4_f32(
      /*neg_a=*/false, a, /*neg_b=*/false, b,
      /*c_mod=*/(short)0, c, /*reuse_a=*/false, /*reuse_b=*/false);
}

// ---------------------------------------------------------------------------
// Kernel 0: zero the workspace accumulators
// ---------------------------------------------------------------------------
__global__ void zero_kernel(float* __restrict__ p, int n) {
  int i = blockIdx.x * 256 + threadIdx.x;
  if (i < n) p[i] = 0.0f;
}

// ---------------------------------------------------------------------------
// Kernel 1: 119 tall-skinny GEMM blocks  G[o,d] = sum_b A[b,o]*B[b,d]
//   grid.x = block id (0..118), grid.y = K-chunk
//   128 threads = 4 wave32; wave w -> 32x32 super-tile (sto = w>>1, std = w&1)
//   2x2 register blocking: 4 WMMAs per 4 fragment b64 loads
// ---------------------------------------------------------------------------
__global__ __launch_bounds__(THREADS) void block_gemm_kernel(
    const float* __restrict__ x1, const float* __restrict__ s1,
    const float* __restrict__ y1, const float* __restrict__ x2,
    const float* __restrict__ s2, const float* __restrict__ y2,
    float* __restrict__ gBlocks) {
  __shared__ float ldsA[64 * LDS_STRIDE];
  __shared__ float ldsB[64 * LDS_STRIDE];

  const int blk   = blockIdx.x;
  const int kbase = blockIdx.y * KCHUNK;
  const int tid   = threadIdx.x;
  const int wave  = tid >> 5;
  const int lane  = tid & 31;
  const int half  = lane >> 4;
  const int mlan  = lane & 15;
  const int sto   = wave >> 1;   // super-tile row (o)   0..1
  const int std_  = wave & 1;    // super-tile col (d)   0..1

  // Uniform (SGPR) per-workgroup exponent/matrix selection
  const float *mA0, *mA1, *mA2, *mA3, *mB0, *mB1;
  int a0, a1, a2, a3, b0, b1;
  if (blk < NW_BLOCKS) {
    mA0 = s1; mA1 = y1; mA2 = s2; mA3 = y2;
    a0 = c_tab.wblkA[blk][0]; a1 = c_tab.wblkA[blk][1];
    a2 = c_tab.wblkA[blk][2]; a3 = c_tab.wblkA[blk][3];
    mB0 = x1; mB1 = x2;
    b0 = c_tab.wblkB[blk][0]; b1 = c_tab.wblkB[blk][1];
  } else {
    const int mb = blk - NW_BLOCKS;
    mA0 = y1; mA1 = y2; mA2 = y1; mA3 = y2;
    a0 = c_tab.mblkA[mb][0]; a1 = c_tab.mblkA[mb][1];
    a2 = 0;                  a3 = 0;
    mB0 = y1; mB1 = y2;
    b0 = c_tab.mblkB[mb][0]; b1 = c_tab.mblkB[mb][1];
  }

  v8f acc00 = {0.f,0.f,0.f,0.f,0.f,0.f,0.f,0.f};
  v8f acc01 = acc00, acc10 = acc00, acc11 = acc00;

  // Fragment base pointers (column-major LDS tiles -> one b64 per fragment half)
  const float* pa0 = ldsA + (32 * sto + mlan) * LDS_STRIDE + 2 * half;
  const float* pa1 = pa0 + 16 * LDS_STRIDE;
  const float* pb0 = ldsB + (32 * std_ + mlan) * LDS_STRIDE + 2 * half;
  const float* pb1 = pb0 + 16 * LDS_STRIDE;

  for (int sub = 0; sub < KCHUNK; sub += KSUB) {
    // ---- stage powered tiles (coalesced global reads, uniform pow branches)
    for (int i = tid; i < KSUB * 64; i += THREADS) {
      const int b = i >> 6;       // row within sub-chunk
      const int c = i & 63;       // column (o or d)
      const int idx = (kbase + sub + b) * 64 + c;
      const float av = powld(mA0, idx, a0) * powld(mA1, idx, a1) *
                       powld(mA2, idx, a2) * powld(mA3, idx, a3);
      const float bv = powld(mB0, idx, b0) * powld(mB1, idx, b1);
      ldsA[c * LDS_STRIDE + b] = av;
      ldsB[c * LDS_STRIDE + b] = bv;
    }
    __syncthreads();

    // ---- 16 k-steps x 4 WMMA (2x2 register blocking) over the sub-chunk
#pragma unroll
    for (int k = 0; k < KSUB; k += 4) {
      v2f af0 = *(const v2f*)(pa0 + k);
      v2f af1 = *(const v2f*)(pa1 + k);
      v2f bf0 = *(const v2f*)(pb0 + k);
      v2f bf1 = *(const v2f*)(pb1 + k);
      acc00 = wmma_f32(af0, bf0, acc00);
      acc01 = wmma_f32(af0, bf1, acc01);
      acc10 = wmma_f32(af1, bf0, acc10);
      acc11 = wmma_f32(af1, bf1, acc11);
    }
    __syncthreads();
  }

  // ---- accumulate this K-chunk's partial into the global block
  float* gout = gBlocks + blk * 4096;
  const int d0 = 32 * std_ + mlan;
  const int o0 = 32 * sto + 8 * half;
#pragma unroll
  for (int j = 0; j < 8; ++j) {
    atomicAdd(gout + (o0 + j) * 64 + (d0), acc00[j]);
    atomicAdd(gout + (o0 + j) * 64 + (d0 + 16), acc01[j]);
    atomicAdd(gout + (o0 + 16 + j) * 64 + (d0), acc10[j]);
    atomicAdd(gout + (o0 + 16 + j) * 64 + (d0 + 16), acc11[j]);
  }
}

// ---------------------------------------------------------------------------
// Kernel 2: column sums  gBsum[c][d] = sum_b y1^e0 * y2^e1
// ---------------------------------------------------------------------------
__global__ __launch_bounds__(256) void colsum_kernel(
    const float* __restrict__ y1, const float* __restrict__ y2,
    float* __restrict__ gBsum) {
  const int tid = threadIdx.x;
  const int d   = tid & 63;
  const int sub = tid >> 6;                     // 0..3
  const int base = blockIdx.x * 1024;           // 64 blocks x 1024 rows
  float s[10];
#pragma unroll
  for (int c = 0; c < 10; ++c) s[c] = 0.0f;

  for (int r = sub; r < 1024; r += 4) {
    const int idx = (base + r) * 64 + d;
    const float a = y1[idx];
    const float b = y2[idx];
    const float pa[4] = {1.0f, a, a * a, a * a * a};
    const float pbv[4] = {1.0f, b, b * b, b * b * b};
#pragma unroll
    for (int c = 0; c < 10; ++c)
      s[c] += pa[c_tab.bcomb[c][0]] * pbv[c_tab.bcomb[c][1]];
  }
#pragma unroll
  for (int c = 0; c < 10; ++c) atomicAdd(gBsum + c * 64 + d, s[c]);
}

// ---------------------------------------------------------------------------
// Kernel 3: finalize — combine blocks with params and elementwise powers
//   out = [delta_w (4096) | delta_b (64) | delta_m (4096)]
// ---------------------------------------------------------------------------
__global__ __launch_bounds__(256) void finalize_kernel(
    const float* __restrict__ w, const float* __restrict__ bvec,
    const float* __restrict__ m, const float* __restrict__ pw,
    const float* __restrict__ pb, const float* __restrict__ pm,
    const float* __restrict__ gBlocks, const float* __restrict__ gBsum,
    float* __restrict__ out) {
  const float invB = 1.0f / 65536.0f;
  const int idx = blockIdx.x * 256 + threadIdx.x;
  if (idx < 4096) {
    const float wv = w[idx];
    const float wp[4] = {1.0f, wv, wv * wv, wv * wv * wv};
    float acc = 0.0f;
    for (int i = 0; i < 120; ++i)
      acc += pw[i] * gBlocks[(int)c_tab.wterm_blk[i] * 4096 + idx] *
             wp[(int)c_tab.wterm_e[i]];
    out[idx] = acc * invB;
  } else if (idx < 4160) {
    const int d = idx - 4096;
    const float bv = bvec[d];
    const float bp[4] = {1.0f, bv, bv * bv, bv * bv * bv};
    float acc = 0.0f;
    for (int i = 0; i < 20; ++i)
      acc += pb[i] * gBsum[(int)c_tab.bterm_c[i] * 64 + d] *
             bp[(int)c_tab.bterm_e[i]];
    out[idx] = acc * invB;
  } else if (idx < 8256) {
    const int j = idx - 4160;
    const float mv = m[j];
    const float mp[4] = {1.0f, mv, mv * mv, mv * mv * mv};
    float acc = 0.0f;
    for (int i = 0; i < 56; ++i)
      acc += pm[i] * gBlocks[(NW_BLOCKS + (int)c_tab.mterm_blk[i]) * 4096 + j] *
             mp[(int)c_tab.mterm_e[i]];
    acc *= invB;
    const int o = j >> 6, d = j & 63;
    if (o == d) acc += pm[56];
    out[idx] = acc;
  }
}

// ---------------------------------------------------------------------------
extern "C" void kernel_launch(void* const* d_in, const int* in_sizes, int n_in,
                              void* d_out, int out_size, void* d_ws, size_t ws_size,
                              hipStream_t stream) {
  (void)in_sizes; (void)n_in; (void)out_size; (void)ws_size;
  const float* x1 = (const float*)d_in[0];
  const float* s1 = (const float*)d_in[1];
  const float* y1 = (const float*)d_in[2];
  const float* x2 = (const float*)d_in[3];
  const float* s2 = (const float*)d_in[4];
  const float* y2 = (const float*)d_in[5];
  const float* w  = (const float*)d_in[6];
  const float* b  = (const float*)d_in[7];
  const float* m  = (const float*)d_in[8];
  const float* pw = (const float*)d_in[9];
  const float* pb = (const float*)d_in[10];
  const float* pm = (const float*)d_in[11];
  float* out = (float*)d_out;

  float* gBlocks = (float*)d_ws;                 // 119 * 4096 floats (~1.95 MB)
  float* gBsum   = gBlocks + N_BLOCKS * 4096;    // 10 * 64 floats

  const int nzero = N_BLOCKS * 4096 + 10 * 64;
  zero_kernel<<<(nzero + 255) / 256, 256, 0, stream>>>(gBlocks, nzero);
  block_gemm_kernel<<<dim3(N_BLOCKS, KCHUNKS), THREADS, 0, stream>>>(
      x1, s1, y1, x2, s2, y2, gBlocks);
  colsum_kernel<<<64, 256, 0, stream>>>(y1, y2, gBsum);
  finalize_kernel<<<(8256 + 255) / 256, 256, 0, stream>>>(
      w, b, m, pw, pb, pm, gBlocks, gBsum, out);
}